// Sparsemax_3212635537843
// MI455X (gfx1250) — compile-verified
//
#include <hip/hip_runtime.h>
#include <hip/hip_bf16.h>

// Sparsemax over rows of 4096 fp32, 8192 rows.
// Memory-bound: 268 MB total traffic -> ~11.5us floor at 23.3 TB/s.
// Michelot projection (no sort): tau <- (sum_{x>tau} x - 1)/|{x>tau}|,
// terminates exactly when the support count stabilizes (<~10 iters typ.).

#define COLS 4096
#define TPB  256
#define V4PT 4   // float4 per thread (16 floats each)

typedef float v4f __attribute__((ext_vector_type(4)));

// wave32 butterfly add via ds_swizzle group-of-32 mode:
// offset = xor_mask<<10 | or_mask<<5 | and_mask(0x1f)
template <int K>
__device__ __forceinline__ float swz_add(float v) {
  const int sw = __builtin_amdgcn_ds_swizzle(__builtin_bit_cast(int, v),
                                             (K << 10) | 0x1f);
  return v + __builtin_bit_cast(float, sw);
}

__device__ __forceinline__ float wave_sum(float v) {
  v = swz_add<16>(v);
  v = swz_add<8>(v);
  v = swz_add<4>(v);
  v = swz_add<2>(v);
  v = swz_add<1>(v);
  return v;
}

__global__ __launch_bounds__(TPB) void sparsemax_kernel(const float* __restrict__ x,
                                                        float* __restrict__ out) {
  __shared__ float  lds_row[COLS];
  __shared__ float2 red[TPB / 32];

  const int tid = threadIdx.x;
  const unsigned long long row_base =
      (unsigned long long)(const void*)(x + (size_t)blockIdx.x * COLS);

  // LDS byte address: low 32 bits of the generic pointer are the LDS offset.
  const unsigned lds_base = (unsigned)(unsigned long long)(uintptr_t)lds_row;
  const unsigned goff  = (unsigned)(tid * 16);
  const unsigned laddr = lds_base + goff;

  // Stage the 16KB row into LDS via CDNA5 async global->LDS copies (ASYNCcnt).
  // INST_OFFSET is applied to both the LDS and global addresses, so one
  // (laddr, goff) pair covers all 4 chunks. Streaming data -> NT hint.
  asm volatile(
      "global_load_async_to_lds_b128 %0, %1, %2 offset:0 th:TH_LOAD_NT\n\t"
      "global_load_async_to_lds_b128 %0, %1, %2 offset:4096 th:TH_LOAD_NT\n\t"
      "global_load_async_to_lds_b128 %0, %1, %2 offset:8192 th:TH_LOAD_NT\n\t"
      "global_load_async_to_lds_b128 %0, %1, %2 offset:12288 th:TH_LOAD_NT"
      :: "v"(laddr), "v"(goff), "s"(row_base)
      : "memory");
  asm volatile("s_wait_asynccnt 0" ::: "memory");
  __syncthreads();

  // Pull my 16 contiguous elements LDS -> registers (ds_load_b128 x4).
  v4f r[V4PT];
  const v4f* lds4 = reinterpret_cast<const v4f*>(lds_row);
#pragma unroll
  for (int j = 0; j < V4PT; ++j)
    r[j] = lds4[tid * V4PT + j];

  const int wave = tid >> 5;
  const int lane = tid & 31;

  // Michelot projection; first pass (tau = -inf) gives tau0 = (sum-1)/n.
  float tau    = -3.4e38f;
  float k_prev = 0.0f;
#pragma unroll 1
  for (int it = 0; it < 24; ++it) {
    float s_loc = 0.f, c_loc = 0.f;
#pragma unroll
    for (int j = 0; j < V4PT; ++j) {
      const v4f v = r[j];
      if (v.x > tau) { s_loc += v.x; c_loc += 1.f; }
      if (v.y > tau) { s_loc += v.y; c_loc += 1.f; }
      if (v.z > tau) { s_loc += v.z; c_loc += 1.f; }
      if (v.w > tau) { s_loc += v.w; c_loc += 1.f; }
    }
    const float s_w = wave_sum(s_loc);
    const float c_w = wave_sum(c_loc);
    if (lane == 0) red[wave] = make_float2(s_w, c_w);
    __syncthreads();
    float S = 0.f, K = 0.f;
#pragma unroll
    for (int w = 0; w < TPB / 32; ++w) {
      const float2 p = red[w];
      S += p.x;
      K += p.y;
    }
    __syncthreads();
    if (K == k_prev) break;          // support stable -> tau exact (uniform branch)
    tau    = (S - 1.0f) / K;
    k_prev = K;
  }

  // out = relu(x - tau), coalesced non-temporal 128-bit stores (streaming).
  v4f* o4 = reinterpret_cast<v4f*>(out + (size_t)blockIdx.x * COLS);
#pragma unroll
  for (int j = 0; j < V4PT; ++j) {
    v4f v = r[j];
    v.x = fmaxf(v.x - tau, 0.f);
    v.y = fmaxf(v.y - tau, 0.f);
    v.z = fmaxf(v.z - tau, 0.f);
    v.w = fmaxf(v.w - tau, 0.f);
    __builtin_nontemporal_store(v, &o4[tid * V4PT + j]);
  }
}

extern "C" void kernel_launch(void* const* d_in, const int* in_sizes, int n_in,
                              void* d_out, int out_size, void* d_ws, size_t ws_size,
                              hipStream_t stream) {
  const float* x = (const float*)d_in[0];
  float* out     = (float*)d_out;
  const int rows = in_sizes[0] / COLS;
  sparsemax_kernel<<<rows, TPB, 0, stream>>>(x, out);
}